// DomainAttention_11312943858092
// MI455X (gfx1250) — compile-verified
//
#include <hip/hip_runtime.h>
#include <hip/hip_bf16.h>
#include <math.h>

// Problem constants from the reference
// x: (B=4, C=256, D=32, H=64, W=64) fp32, GROUPS=4 -> Dg=8
// NDS=3 datasets, RED=16 -> r=16
// M dimension for WMMA = B*GROUPS = 16 (exact 16x16 tile rows)
#define BATCH   4
#define PLANES  256
#define DEPTH   32
#define GRP     4
#define DG      8          // DEPTH/GRP
#define HW      4096       // 64*64
#define NDS     3
#define RDIM    16         // PLANES/RED
#define CHUNK   32768      // DG*HW floats, contiguous per (b,c,g)

typedef float v2f __attribute__((ext_vector_type(2)));
typedef float v8f __attribute__((ext_vector_type(8)));

__device__ __forceinline__ v8f wmma_f32_16x16x4(v2f a, v2f b, v8f c) {
  // (neg_a, A, neg_b, B, c_mod, C, reuse_a, reuse_b)
  return __builtin_amdgcn_wmma_f32_16x16x4_f32(false, a, false, b, (short)0, c,
                                               false, false);
}

// ---------------------------------------------------------------------------
// Kernel A: group-wise global average pool.
// One block per (b, c, g) slice; slice is a contiguous CHUNK of 32768 floats.
// st layout: st[(b*GRP + g)*PLANES + c]  (the WMMA "M" rows are bg = b*4+g)
// ---------------------------------------------------------------------------
__global__ __launch_bounds__(256) void pool_kernel(const float* __restrict__ x,
                                                   float* __restrict__ st) {
  __shared__ float red[8];
  const int bid = blockIdx.x;              // = b*1024 + c*4 + g
  const int t = threadIdx.x;
  const int g = bid & 3;
  const int c = (bid >> 2) & 255;
  const int b = bid >> 10;

  const float4* p = (const float4*)(x + (size_t)bid * CHUNK);
  float s = 0.f;
#pragma unroll
  for (int i = 0; i < 32; ++i) {           // 256 thr * 32 float4 = 32768 floats
    float4 v = p[t + i * 256];
    s += v.x + v.y + v.z + v.w;
  }
  // wave32 reduce, then cross-wave via LDS
  for (int off = 16; off > 0; off >>= 1) s += __shfl_down(s, off, 32);
  if ((t & 31) == 0) red[t >> 5] = s;
  __syncthreads();
  if (t == 0) {
    float tot = 0.f;
#pragma unroll
    for (int i = 0; i < 8; ++i) tot += red[i];
    st[(b * GRP + g) * PLANES + c] = tot * (1.0f / (float)CHUNK);
  }
}

// ---------------------------------------------------------------------------
// Kernel B: per-dataset SE excitation + softmax routing + sigmoid gate.
// Single wave32. M = 16 (b,g) rows.
//   GEMM1 (per n): H_n = relu(ST(16x256) @ w1_n^T(256x16) + b1_n)   [WMMA]
//   logits       = ST @ wf^T + bf ; wgt = softmax over n            [scalar]
//   GEMM2 (per n): Y_n = H_n(16x16) @ w2_n^T(16x256) + b2_n          [WMMA]
//   gate = sigmoid(sum_n Y_n * wgt[:,n])
//
// f32 WMMA VGPR layouts (ISA 7.12.2):
//   A 16x4 : lane l -> row (l&15); VGPR0 = K base+{0 | 2}, VGPR1 = +1
//   B 4x16 : lane l -> col (l&15); VGPR0 = K base+{0 | 2}, VGPR1 = +1
//   C 16x16: VGPR i, lane l -> row i + (l<16 ? 0 : 8), col (l&15)
// ---------------------------------------------------------------------------
__global__ __launch_bounds__(32) void se_gate_kernel(
    const float* __restrict__ st,   // [16][256]
    const float* __restrict__ w1,   // [3][16][256]  (n, r, c)
    const float* __restrict__ b1,   // [3][16]
    const float* __restrict__ w2,   // [3][256][16]  (n, c, r)
    const float* __restrict__ b2,   // [3][256]
    const float* __restrict__ wf,   // [3][256]
    const float* __restrict__ bf,   // [3]
    float* __restrict__ gate)       // [16][256]
{
  __shared__ float st_lds[16 * PLANES];
  __shared__ float h_lds[NDS][16][RDIM];
  __shared__ float logit_lds[16][NDS];
  __shared__ float wgt_lds[16][4];

  const int lane = threadIdx.x;
  const int col  = lane & 15;               // row for A, col for B/C halves
  const int koff = (lane < 16) ? 0 : 2;
  const int mhal = (lane < 16) ? 0 : 8;     // C row offset per lane half

  // stage descriptor into LDS
  for (int i = 0; i < 128; ++i) st_lds[lane + i * 32] = st[lane + i * 32];
  __syncthreads();

  // ---- GEMM1: H_n = relu(ST @ w1_n^T + b1_n), 3 x (16x16x256) -------------
  for (int n = 0; n < NDS; ++n) {
    v8f acc;
#pragma unroll
    for (int i = 0; i < 8; ++i) acc[i] = 0.f;
    const float* w1row = w1 + (n * RDIM + col) * PLANES;  // B col = r index
    const float* strow = st_lds + col * PLANES;           // A row = bg index
    for (int kb = 0; kb < PLANES; kb += 4) {
      v2f a, b;
      a.x = strow[kb + koff];     a.y = strow[kb + koff + 1];
      b.x = w1row[kb + koff];     b.y = w1row[kb + koff + 1];
      acc = wmma_f32_16x16x4(a, b, acc);
    }
    const float bias = b1[n * RDIM + col];
#pragma unroll
    for (int i = 0; i < 8; ++i) {
      float v = acc[i] + bias;
      h_lds[n][i + mhal][col] = v > 0.f ? v : 0.f;
    }
  }
  __syncthreads();

  // ---- routing logits + softmax (48 dot products of length 256) -----------
  for (int p = lane; p < 16 * NDS; p += 32) {
    const int m = p / NDS, n = p % NDS;
    const float* wfn = wf + n * PLANES;
    const float* stm = st_lds + m * PLANES;
    float s = bf[n];
    for (int k = 0; k < PLANES; ++k) s = fmaf(stm[k], wfn[k], s);
    logit_lds[m][n] = s;
  }
  __syncthreads();
  if (lane < 16) {
    float l0 = logit_lds[lane][0], l1 = logit_lds[lane][1], l2 = logit_lds[lane][2];
    float mx = fmaxf(l0, fmaxf(l1, l2));
    float e0 = expf(l0 - mx), e1 = expf(l1 - mx), e2 = expf(l2 - mx);
    float inv = 1.f / (e0 + e1 + e2);
    wgt_lds[lane][0] = e0 * inv;
    wgt_lds[lane][1] = e1 * inv;
    wgt_lds[lane][2] = e2 * inv;
  }
  __syncthreads();

  // per-lane routing weights in C-layout order (row = i + mhal)
  float wv[NDS][8];
#pragma unroll
  for (int n = 0; n < NDS; ++n)
#pragma unroll
    for (int i = 0; i < 8; ++i) wv[n][i] = wgt_lds[i + mhal][n];

  // ---- GEMM2 + weighted combine + sigmoid, 16 N-tiles of 16 channels ------
  for (int t = 0; t < 16; ++t) {
    float gacc[8];
#pragma unroll
    for (int i = 0; i < 8; ++i) gacc[i] = 0.f;
    for (int n = 0; n < NDS; ++n) {
      v8f acc;
#pragma unroll
      for (int i = 0; i < 8; ++i) acc[i] = 0.f;
      const float* hrow  = &h_lds[n][col][0];                     // A row = bg
      const float* w2row = w2 + (n * PLANES + t * 16 + col) * RDIM; // B col = c
#pragma unroll
      for (int kb = 0; kb < RDIM; kb += 4) {
        v2f a, b;
        a.x = hrow[kb + koff];    a.y = hrow[kb + koff + 1];
        b.x = w2row[kb + koff];   b.y = w2row[kb + koff + 1];
        acc = wmma_f32_16x16x4(a, b, acc);
      }
      const float bias = b2[n * PLANES + t * 16 + col];
#pragma unroll
      for (int i = 0; i < 8; ++i) gacc[i] += (acc[i] + bias) * wv[n][i];
    }
#pragma unroll
    for (int i = 0; i < 8; ++i) {
      float g = 1.f / (1.f + expf(-gacc[i]));
      gate[(i + mhal) * PLANES + t * 16 + col] = g;
    }
  }
}

// ---------------------------------------------------------------------------
// Kernel C: out = x * gate[b][g][c]. 4 blocks per contiguous (b,c,g) chunk,
// 256 threads x 8 float4 each. gate lookup is block-uniform (scalar load).
// ---------------------------------------------------------------------------
__global__ __launch_bounds__(256) void apply_kernel(const float* __restrict__ x,
                                                    const float* __restrict__ gate,
                                                    float* __restrict__ out) {
  const int bid = blockIdx.x;
  const int chunkId = bid >> 2;            // = b*1024 + c*4 + g
  const int sub = bid & 3;
  const int g = chunkId & 3;
  const int c = (chunkId >> 2) & 255;
  const int b = chunkId >> 10;
  const float gs = gate[(b * GRP + g) * PLANES + c];

  const size_t base4 = (size_t)chunkId * (CHUNK / 4) + (size_t)sub * 2048;
  const float4* pin = (const float4*)x + base4;
  float4* pout = (float4*)out + base4;
  const int t = threadIdx.x;
#pragma unroll
  for (int i = 0; i < 8; ++i) {
    float4 v = pin[t + i * 256];
    v.x *= gs; v.y *= gs; v.z *= gs; v.w *= gs;
    pout[t + i * 256] = v;
  }
}

extern "C" void kernel_launch(void* const* d_in, const int* in_sizes, int n_in,
                              void* d_out, int out_size, void* d_ws, size_t ws_size,
                              hipStream_t stream) {
  const float* x  = (const float*)d_in[0];
  const float* w1 = (const float*)d_in[1];
  const float* b1 = (const float*)d_in[2];
  const float* w2 = (const float*)d_in[3];
  const float* b2 = (const float*)d_in[4];
  const float* wf = (const float*)d_in[5];
  const float* bf = (const float*)d_in[6];
  float* out  = (float*)d_out;
  float* st   = (float*)d_ws;        // 16*256 floats
  float* gate = st + 16 * PLANES;    // 16*256 floats

  // Pass 1: read x once (512 MB) -> pooled descriptors
  pool_kernel<<<BATCH * PLANES * GRP, 256, 0, stream>>>(x, st);
  // Tiny SE/routing math on one wave, WMMA f32 16x16x4
  se_gate_kernel<<<1, 32, 0, stream>>>(st, w1, b1, w2, b2, wf, bf, gate);
  // Pass 2: read x + write out (1 GB)
  apply_kernel<<<(BATCH * PLANES * GRP) * 4, 256, 0, stream>>>(x, gate, out);
}